// ElasticTransformLayer_65283502899753
// MI455X (gfx1250) — compile-verified
//
#include <hip/hip_runtime.h>
#include <hip/hip_bf16.h>
#include <math.h>

// Elastic transform for MI455X (gfx1250, wave32).
// Spline prefilter GEMMs run as *banded* f32 WMMA (v_wmma_f32_16x16x4_f32):
// inverse collocation entries decay as 0.268^d -> half-bandwidth 24 is exact
// to < 1e-13, so each 16-row tile only needs a 64-wide K window (16 WMMAs).
// All K-loop loads/stores use one base pointer + constant immediate offsets
// so the unrolled loop is pure global_load(offset:N) + v_wmma.

#define BN 32
#define HH 256
#define WW 1024
#define CC 3
#define WC (WW * CC)        // 3072
#define HB 24               // half bandwidth of the banded inverse
#define KW 64               // K window per 16-row tile (16 wmma steps)

typedef float v2f __attribute__((ext_vector_type(2)));
typedef float v8f __attribute__((ext_vector_type(8)));

// ---------------- helpers ----------------

__device__ __forceinline__ float b3f(float t) {
    float a = fabsf(t);
    if (a < 1.0f) return (4.0f - 6.0f * a * a + 3.0f * a * a * a) * (1.0f / 6.0f);
    if (a < 2.0f) { float u = 2.0f - a; return u * u * u * (1.0f / 6.0f); }
    return 0.0f;
}

// cubic B-spline basis over a 3-point control grid, mirror boundary (m = 4)
__device__ __forceinline__ void basis3(float u, float* w) {
    w[0] = 0.0f; w[1] = 0.0f; w[2] = 0.0f;
    int base = (int)floorf(u);
    #pragma unroll
    for (int k = -1; k < 3; ++k) {
        int idx = base + k;
        int j = abs(idx) & 3;           // abs(idx) % 4
        j = min(j, 4 - j);
        if (j < 3) w[j] += b3f(u - (float)idx);
    }
}

// ---------------- setup: tridiagonal inverse via Thomas per column ----------------
// A: diag 2/3; sub/super 1/6 interior; A[0,1]=A[n-1,n-2]=1/3 (mirror fold).
// Thread j solves A x = e_j -> column j of A^{-1}.
__global__ void prep_tridiag_inv(float* __restrict__ out, int n) {
    __shared__ float cp[1024];
    if (threadIdx.x == 0) {
        cp[0] = (1.0f / 3.0f) / (2.0f / 3.0f);
        for (int i = 1; i < n; ++i) {
            float a = (i == n - 1) ? (1.0f / 3.0f) : (1.0f / 6.0f);
            float c = (i == n - 1) ? 0.0f : (1.0f / 6.0f);
            float m = (2.0f / 3.0f) - a * cp[i - 1];
            cp[i] = c / m;
        }
    }
    __syncthreads();
    int j = threadIdx.x;
    if (j < n) {
        // forward sweep; dp stored in out column j
        float prev = 0.0f;
        for (int i = 0; i < n; ++i) {
            float d = (i == j) ? 1.0f : 0.0f;
            float m, a;
            if (i == 0) { m = 2.0f / 3.0f; a = 0.0f; }
            else {
                a = (i == n - 1) ? (1.0f / 3.0f) : (1.0f / 6.0f);
                m = (2.0f / 3.0f) - a * cp[i - 1];
            }
            prev = (d - a * prev) / m;
            out[(size_t)i * n + j] = prev;
        }
        // back substitution in place
        float x = out[(size_t)(n - 1) * n + j];
        for (int i = n - 2; i >= 0; --i) {
            x = out[(size_t)i * n + j] - cp[i] * x;
            out[(size_t)i * n + j] = x;
        }
    }
}

// ---------------- setup: dense displacement fields from 3x3 control grid ----------------
__global__ void prep_field(const float* __restrict__ disp,
                           float* __restrict__ dy, float* __restrict__ dx) {
    int idx = blockIdx.x * blockDim.x + threadIdx.x;
    if (idx >= HH * WW) return;
    int w = idx % WW, h = idx / WW;

    // A3 (3x3 collocation) and its closed-form inverse
    const float A00 = 2.f/3.f, A01 = 1.f/3.f, A02 = 0.f;
    const float A10 = 1.f/6.f, A11 = 2.f/3.f, A12 = 1.f/6.f;
    const float A20 = 0.f,     A21 = 1.f/3.f, A22 = 2.f/3.f;
    float det = A00*(A11*A22 - A12*A21) - A01*(A10*A22 - A12*A20) + A02*(A10*A21 - A11*A20);
    float id = 1.0f / det;
    float inv[3][3];
    inv[0][0] =  (A11*A22 - A12*A21)*id; inv[0][1] = -(A01*A22 - A02*A21)*id; inv[0][2] =  (A01*A12 - A02*A11)*id;
    inv[1][0] = -(A10*A22 - A12*A20)*id; inv[1][1] =  (A00*A22 - A02*A20)*id; inv[1][2] = -(A00*A12 - A02*A10)*id;
    inv[2][0] =  (A10*A21 - A11*A20)*id; inv[2][1] = -(A00*A21 - A01*A20)*id; inv[2][2] =  (A00*A11 - A01*A10)*id;

    // Cy = inv * (5*D0) * inv^T ; Cx = inv * (5*D1) * inv^T
    float Cy[3][3], Cx[3][3];
    #pragma unroll
    for (int i = 0; i < 3; ++i) {
        float t0[3], t1[3];
        #pragma unroll
        for (int jj = 0; jj < 3; ++jj) {
            t0[jj] = 0.f; t1[jj] = 0.f;
            #pragma unroll
            for (int a = 0; a < 3; ++a) {
                t0[jj] += inv[i][a] * (5.0f * disp[0 * 9 + a * 3 + jj]);
                t1[jj] += inv[i][a] * (5.0f * disp[1 * 9 + a * 3 + jj]);
            }
        }
        #pragma unroll
        for (int jj = 0; jj < 3; ++jj) {
            float s0 = 0.f, s1 = 0.f;
            #pragma unroll
            for (int bsum = 0; bsum < 3; ++bsum) {
                s0 += t0[bsum] * inv[jj][bsum];
                s1 += t1[bsum] * inv[jj][bsum];
            }
            Cy[i][jj] = s0; Cx[i][jj] = s1;
        }
    }

    float wh[3], ww2[3];
    basis3((float)h * 2.0f / (float)(HH - 1), wh);
    basis3((float)w * 2.0f / (float)(WW - 1), ww2);
    float sy = 0.f, sx = 0.f;
    #pragma unroll
    for (int i = 0; i < 3; ++i)
        #pragma unroll
        for (int jj = 0; jj < 3; ++jj) {
            float wgt = wh[i] * ww2[jj];
            sy += wgt * Cy[i][jj];
            sx += wgt * Cx[i][jj];
        }
    dy[idx] = sy; dx[idx] = sx;
}

// ---------------- banded WMMA GEMM along H: tmp[b] = AHinv @ x[b] ----------------
// M = 256 (p), K = 256 (h, banded 64), N = 3072 (w*3+c). One wave per 16x16 tile.
__global__ __launch_bounds__(128) void gemm_h(const float* __restrict__ Ainv,
                                              const float* __restrict__ x,
                                              float* __restrict__ tmp) {
    int wave = threadIdx.x >> 5, lane = threadIdx.x & 31;
    int nblk = blockIdx.x * 4 + wave;          // 0..191
    int p0 = blockIdx.y * 16;                  // 0..240
    int b  = blockIdx.z;
    const int Kdim = HH;

    int col   = nblk * 16 + (lane & 15);       // N index
    int row   = p0 + (lane & 15);              // A row for this lane
    int khalf = (lane >> 4) << 1;              // 0 or 2 (A/B K sub-lane split)
    int kstart = p0 - HB;
    if (kstart < 0) kstart = 0;
    if (kstart > Kdim - KW) kstart = Kdim - KW;
    int kbase = kstart + khalf;                // even -> 8B-aligned A loads

    // single base pointers; all K-loop accesses become constant immediate offsets
    const float* ap = Ainv + row * Kdim + kbase;
    const float* bp = x + (size_t)b * Kdim * WC + (size_t)kbase * WC + col;

    v8f acc = {};
    #pragma unroll
    for (int kk = 0; kk < KW; kk += 4) {
        v2f a = *(const v2f*)(ap + kk);        // A[row, k], A[row, k+1]
        v2f bb;
        bb.x = bp[kk * WC];                    // B[k,   col]
        bb.y = bp[kk * WC + WC];               // B[k+1, col]
        acc = __builtin_amdgcn_wmma_f32_16x16x4_f32(false, a, false, bb,
                                                    (short)0, acc, false, false);
    }
    // D layout: VGPR r, lanes 0-15 -> M=r; lanes 16-31 -> M=8+r; N = lane&15
    int mbase = p0 + ((lane >> 4) << 3);
    float* op = tmp + (size_t)b * Kdim * WC + (size_t)mbase * WC + col;
    #pragma unroll
    for (int r = 0; r < 8; ++r)
        op[r * WC] = acc[r];
}

// ---------------- banded WMMA GEMM along W: coef[b,p,q,c] = sum_w AWinv[q,w] tmp[b,p,w,c] ----
// M = 1024 (q), K = 1024 (w, banded 64), N = 768 (p*3+c).
__global__ __launch_bounds__(128) void gemm_w(const float* __restrict__ Ainv,
                                              const float* __restrict__ tmp,
                                              float* __restrict__ coef) {
    int wave = threadIdx.x >> 5, lane = threadIdx.x & 31;
    int nblk = blockIdx.x * 4 + wave;          // 0..47
    int q0 = blockIdx.y * 16;                  // 0..1008
    int b  = blockIdx.z;
    const int Kdim = WW;

    int n = nblk * 16 + (lane & 15);           // 0..767
    int p = n / 3, c = n - 3 * p;
    int row   = q0 + (lane & 15);
    int khalf = (lane >> 4) << 1;
    int kstart = q0 - HB;
    if (kstart < 0) kstart = 0;
    if (kstart > Kdim - KW) kstart = Kdim - KW;
    int kbase = kstart + khalf;                // even

    const float* ap = Ainv + (size_t)row * Kdim + kbase;
    const float* bp = tmp + (size_t)b * HH * WC + (size_t)p * WC + c + (size_t)kbase * CC;

    v8f acc = {};
    #pragma unroll
    for (int kk = 0; kk < KW; kk += 4) {
        v2f a = *(const v2f*)(ap + kk);
        v2f bb;
        bb.x = bp[kk * CC];                    // tmp[b,p,k,  c]
        bb.y = bp[kk * CC + CC];               // tmp[b,p,k+1,c]
        acc = __builtin_amdgcn_wmma_f32_16x16x4_f32(false, a, false, bb,
                                                    (short)0, acc, false, false);
    }
    int mbase = q0 + ((lane >> 4) << 3);
    float* op = coef + (size_t)b * HH * WC + (size_t)p * WC + (size_t)mbase * CC + c;
    #pragma unroll
    for (int r = 0; r < 8; ++r)
        op[r * CC] = acc[r];
}

// ---------------- final 4x4 cubic B-spline evaluation ----------------
__global__ void sample_k(const float* __restrict__ coef, const float* __restrict__ dy,
                         const float* __restrict__ dx, float* __restrict__ out) {
    int idx = blockIdx.x * blockDim.x + threadIdx.x;    // over B*H*W
    if (idx >= BN * HH * WW) return;
    int w = idx % WW; int t = idx / WW; int h = t % HH; int b = t / HH;

    float cy = (float)h + dy[h * WW + w];
    float cx = (float)w + dx[h * WW + w];
    int by = (int)floorf(cy), bx = (int)floorf(cx);

    float wxv[4]; int ixv[4];
    #pragma unroll
    for (int kx = 0; kx < 4; ++kx) {
        int i = bx + kx - 1;
        ixv[kx] = min(max(i, 0), WW - 1);
        wxv[kx] = b3f(cx - (float)i);
    }

    float a0 = 0.f, a1 = 0.f, a2 = 0.f;
    const float* cbase = coef + (size_t)b * HH * WC;
    #pragma unroll
    for (int ky = 0; ky < 4; ++ky) {
        int iyr = by + ky - 1;
        int iy = min(max(iyr, 0), HH - 1);
        float wy = b3f(cy - (float)iyr);
        const float* rowp = cbase + (size_t)iy * WC;
        #pragma unroll
        for (int kx = 0; kx < 4; ++kx) {
            float wgt = wy * wxv[kx];
            const float* pv = rowp + ixv[kx] * CC;
            a0 += wgt * pv[0];
            a1 += wgt * pv[1];
            a2 += wgt * pv[2];
        }
    }
    float* o = out + (size_t)idx * CC;
    o[0] = fminf(fmaxf(a0, 0.f), 1.f);
    o[1] = fminf(fmaxf(a1, 0.f), 1.f);
    o[2] = fminf(fmaxf(a2, 0.f), 1.f);
}

// ---------------- launcher ----------------
extern "C" void kernel_launch(void* const* d_in, const int* in_sizes, int n_in,
                              void* d_out, int out_size, void* d_ws, size_t ws_size,
                              hipStream_t stream) {
    const float* x    = (const float*)d_in[0];   // [32,256,1024,3]
    const float* disp = (const float*)d_in[1];   // [2,3,3]
    float* out = (float*)d_out;
    float* ws  = (float*)d_ws;

    // workspace layout (floats): AHinv | AWinv | dy | dx | coef  (~107 MB)
    float* AHinv = ws;
    float* AWinv = AHinv + (size_t)HH * HH;
    float* dy    = AWinv + (size_t)WW * WW;
    float* dx    = dy + (size_t)HH * WW;
    float* coef  = dx + (size_t)HH * WW;
    float* tmp   = out;   // reuse d_out as GEMM1->GEMM2 scratch (overwritten later)

    prep_tridiag_inv<<<1, HH, 0, stream>>>(AHinv, HH);
    prep_tridiag_inv<<<1, WW, 0, stream>>>(AWinv, WW);
    prep_field<<<(HH * WW + 255) / 256, 256, 0, stream>>>(disp, dy, dx);

    dim3 g1(WC / 16 / 4, HH / 16, BN);     // (48, 16, 32)
    gemm_h<<<g1, 128, 0, stream>>>(AHinv, x, tmp);

    dim3 g2(HH * CC / 16 / 4, WW / 16, BN); // (12, 64, 32)
    gemm_w<<<g2, 128, 0, stream>>>(AWinv, tmp, coef);

    sample_k<<<(BN * HH * WW + 255) / 256, 256, 0, stream>>>(coef, dy, dx, out);
}